// MethodDeepGCNResNet_84945863180848
// MI455X (gfx1250) — compile-verified
//
#include <hip/hip_runtime.h>
#include <hip/hip_bf16.h>
#include <math.h>

typedef __attribute__((ext_vector_type(2))) float v2f;
typedef __attribute__((ext_vector_type(8))) float v8f;

#define N_NODES 50000
#define N_EDGES 800000
#define F_IN    256
#define HID     256
#define C_OUT   40

#define GEMM_K   256                 // every GEMM in this net has K = 256
#define LDS_PADP 80                  // float2 stride per k-pair row (bank-conflict-free)

// ---------------------------------------------------------------------------
// Repack weights: Wt[Npad x K] (column-major view of W, zero-padded columns)
//   Wt[n*K + k] = (n < N) ? W[k*N + n] : 0
// ---------------------------------------------------------------------------
__global__ void transpose_pad(const float* __restrict__ W,
                              float* __restrict__ Wt,
                              int K, int N, int Npad) {
    int id = blockIdx.x * blockDim.x + threadIdx.x;
    if (id >= Npad * K) return;
    int n = id / K;
    int k = id - n * K;
    Wt[id] = (n < N) ? W[(size_t)k * N + n] : 0.f;
}

// ---------------------------------------------------------------------------
// Dense GEMM via fp32 WMMA: C[M x N] = A[M x K=256] @ W, W given as padded
// transposed Bt[Npad x 256].
//
// Block = 256 threads = 8 waves. grid.x = ceil(M/16/8), grid.y = Npad/64.
// All waves of a block share one 64-column group: the Bt slab (64 x 256 fp32)
// is cooperatively staged into LDS once (pair-major, stride-80 float2 rows ->
// ds_load_b64 fragment reads are bank-conflict-free), then each wave runs an
// independent 16x64 tile: per k-step 1 pipelined global v2f A-load,
// 4 ds_load_b64 B-frags, 4 v_wmma_f32_16x16x4_f32.
// ---------------------------------------------------------------------------
__global__ void gemm_wmma_f32(const float* __restrict__ A,
                              const float* __restrict__ Bt,
                              float* __restrict__ C,
                              int M, int N) {
    extern __shared__ v2f bs[];                       // [128][LDS_PADP] float2
    constexpr int K  = GEMM_K;
    constexpr int KP = GEMM_K / 2;                    // 128 k-pairs

    const int tid  = threadIdx.x;
    const int lane = tid & 31;
    const int wave = tid >> 5;
    const int n0   = blockIdx.y * 64;

    // ---- cooperative stage of Bt slab: 64 cols x 128 pairs ----
    // e = n*KP + p : consecutive tids -> consecutive pairs of one column
    // (coalesced float2 reads); write bs[p*LDS_PADP + n].
    for (int e = tid; e < 64 * KP; e += 256) {
        int n = e >> 7;                               // e / KP   (KP = 128)
        int p = e & (KP - 1);                         // e % KP
        bs[p * LDS_PADP + n] = *(const v2f*)(Bt + (size_t)(n0 + n) * K + 2 * p);
    }
    __syncthreads();

    const int m0 = (blockIdx.x * 8 + wave) * 16;
    if (m0 >= M) return;                              // tail waves: fill-only

    const int half = lane >> 4;                       // 0: lanes 0-15, 1: 16-31
    const int l16  = lane & 15;

    v8f acc0 = {}, acc1 = {}, acc2 = {}, acc3 = {};

    const float* arow = A + (size_t)(m0 + l16) * K + half * 2;
    const v2f*   brow = bs + half * LDS_PADP + l16;   // + (k/2)*LDS_PADP + tile*16

    v2f a = *(const v2f*)(arow);                      // k = 0 fragment

    #pragma unroll 4
    for (int k = 0; k < K; k += 4) {
        // prefetch next A fragment (overlaps the WMMA batch below)
        const int kn = (k + 4 < K) ? (k + 4) : k;
        v2f a_nxt = *(const v2f*)(arow + kn);

        const v2f* bk = brow + (k >> 1) * LDS_PADP;
        v2f b0 = bk[0];
        v2f b1 = bk[16];
        v2f b2 = bk[32];
        v2f b3 = bk[48];

        acc0 = __builtin_amdgcn_wmma_f32_16x16x4_f32(false, a, false, b0, (short)0, acc0, false, false);
        acc1 = __builtin_amdgcn_wmma_f32_16x16x4_f32(false, a, false, b1, (short)0, acc1, false, false);
        acc2 = __builtin_amdgcn_wmma_f32_16x16x4_f32(false, a, false, b2, (short)0, acc2, false, false);
        acc3 = __builtin_amdgcn_wmma_f32_16x16x4_f32(false, a, false, b3, (short)0, acc3, false, false);
        a = a_nxt;
    }

    // Store: VGPR r holds M = m0 + r + half*8, column = n_tile + l16.
    const int n_a = n0 + 0  + l16;
    const int n_b = n0 + 16 + l16;
    const int n_c = n0 + 32 + l16;
    const int n_d = n0 + 48 + l16;
    const bool va = n_a < N, vb = n_b < N, vc = n_c < N, vd = n_d < N;

    #pragma unroll
    for (int r = 0; r < 8; ++r) {
        float* crow = C + (size_t)(m0 + r + half * 8) * N;
        if (va) crow[n_a] = acc0[r];
        if (vb) crow[n_b] = acc1[r];
        if (vc) crow[n_c] = acc2[r];
        if (vd) crow[n_d] = acc3[r];
    }
}

// ---------------------------------------------------------------------------
// Zero-fill (float4 granularity)
// ---------------------------------------------------------------------------
__global__ void zero_f32x4(float4* __restrict__ p, long long n4) {
    long long i = (long long)blockIdx.x * blockDim.x + threadIdx.x;
    if (i < n4) p[i] = make_float4(0.f, 0.f, 0.f, 0.f);
}

// ---------------------------------------------------------------------------
// SpMM scatter: Y[row[e], :] += val[e] * X[col[e], :]
// One thread per (edge, float4 chunk); CHUNKS compile-time -> shift/mul math.
// ---------------------------------------------------------------------------
template <int CHUNKS>
__global__ void spmm_scatter(const int* __restrict__ erow,
                             const int* __restrict__ ecol,
                             const float* __restrict__ eval,
                             const float* __restrict__ X,
                             float* __restrict__ Y,
                             int nEdges) {
    constexpr int F = CHUNKS * 4;
    long long id = (long long)blockIdx.x * blockDim.x + threadIdx.x;
    long long total = (long long)nEdges * CHUNKS;
    if (id >= total) return;
    int e = (int)(id / CHUNKS);
    int c = (int)(id - (long long)e * CHUNKS);
    int f = c * 4;
    int s = ecol[e];
    int d = erow[e];
    float v = eval[e];
    float4 x = *(const float4*)(X + (size_t)s * F + f);
    float* yp = Y + (size_t)d * F + f;
    unsafeAtomicAdd(yp + 0, v * x.x);
    unsafeAtomicAdd(yp + 1, v * x.y);
    unsafeAtomicAdd(yp + 2, v * x.z);
    unsafeAtomicAdd(yp + 3, v * x.w);
}

// ---------------------------------------------------------------------------
// x = relu(x + res) elementwise, float4 granularity
// ---------------------------------------------------------------------------
__global__ void add_res_relu(float4* __restrict__ x,
                             const float4* __restrict__ res, long long n4) {
    long long i = (long long)blockIdx.x * blockDim.x + threadIdx.x;
    if (i >= n4) return;
    float4 a = x[i];
    float4 b = res[i];
    a.x = fmaxf(a.x + b.x, 0.f);
    a.y = fmaxf(a.y + b.y, 0.f);
    a.z = fmaxf(a.z + b.z, 0.f);
    a.w = fmaxf(a.w + b.w, 0.f);
    x[i] = a;
}

// ---------------------------------------------------------------------------
// out = log_softmax(ya + yb) per row; one wave per row, C <= 64
// ---------------------------------------------------------------------------
__global__ void log_softmax_rows(const float* __restrict__ ya,
                                 const float* __restrict__ yb,
                                 float* __restrict__ out, int nRows, int C) {
    int gwave = (int)((blockIdx.x * (long long)blockDim.x + threadIdx.x) >> 5);
    int lane  = threadIdx.x & 31;
    if (gwave >= nRows) return;

    const float* a = ya + (size_t)gwave * C;
    const float* b = yb + (size_t)gwave * C;

    float v0 = (lane      < C) ? a[lane]      + b[lane]      : -INFINITY;
    float v1 = (lane + 32 < C) ? a[lane + 32] + b[lane + 32] : -INFINITY;

    float m = fmaxf(v0, v1);
    #pragma unroll
    for (int o = 16; o > 0; o >>= 1) m = fmaxf(m, __shfl_xor(m, o, 32));

    float s = 0.f;
    if (lane      < C) s += expf(v0 - m);
    if (lane + 32 < C) s += expf(v1 - m);
    #pragma unroll
    for (int o = 16; o > 0; o >>= 1) s += __shfl_xor(s, o, 32);

    float ls = logf(s);
    float* orow = out + (size_t)gwave * C;
    if (lane      < C) orow[lane]      = v0 - m - ls;
    if (lane + 32 < C) orow[lane + 32] = v1 - m - ls;
}

// ---------------------------------------------------------------------------
// Host-side launcher
// ---------------------------------------------------------------------------
extern "C" void kernel_launch(void* const* d_in, const int* in_sizes, int n_in,
                              void* d_out, int out_size, void* d_ws, size_t ws_size,
                              hipStream_t stream) {
    const float* raw_x      = (const float*)d_in[0];
    const int*   edge_row   = (const int*)  d_in[1];
    const int*   edge_col   = (const int*)  d_in[2];
    const float* edge_val   = (const float*)d_in[3];
    const float* W_gc[3]    = { (const float*)d_in[4], (const float*)d_in[5], (const float*)d_in[6] };
    const float* W_gc3      = (const float*)d_in[7];
    const float* W_res0     = (const float*)d_in[8];
    const float* W_res_last = (const float*)d_in[9];

    const int M = N_NODES, K = GEMM_K, E = N_EDGES;
    const int CPAD = 64;                          // C_OUT padded to 64
    const size_t matElems = (size_t)M * HID;      // 12.8M floats

    // Workspace layout
    float* res0  = (float*)d_ws;                  // [M, HID]
    float* bufA  = res0 + matElems;               // [M, HID]
    float* bufB  = bufA + matElems;               // [M, HID]
    float* wtRes = bufB + matElems;               // [HID,  K]
    float* wtGC0 = wtRes + (size_t)HID * K;
    float* wtGC1 = wtGC0 + (size_t)HID * K;
    float* wtGC2 = wtGC1 + (size_t)HID * K;
    float* wtGC3 = wtGC2 + (size_t)HID * K;       // [CPAD, K]
    float* wtRL  = wtGC3 + (size_t)CPAD * K;      // [CPAD, K]

    // --- repack all weights (padded transpose) ---
    {
        const int tH = HID * K, tC = CPAD * K;
        const int bH = (tH + 255) / 256, bC = (tC + 255) / 256;
        transpose_pad<<<bH, 256, 0, stream>>>(W_res0,     wtRes, K, HID,   HID);
        transpose_pad<<<bH, 256, 0, stream>>>(W_gc[0],    wtGC0, K, HID,   HID);
        transpose_pad<<<bH, 256, 0, stream>>>(W_gc[1],    wtGC1, K, HID,   HID);
        transpose_pad<<<bH, 256, 0, stream>>>(W_gc[2],    wtGC2, K, HID,   HID);
        transpose_pad<<<bC, 256, 0, stream>>>(W_gc3,      wtGC3, K, C_OUT, CPAD);
        transpose_pad<<<bC, 256, 0, stream>>>(W_res_last, wtRL,  K, C_OUT, CPAD);
    }

    const dim3 gemmBlk(256);                                  // 8 waves
    const int  gemmGrdX = (M / 16 + 7) / 8;                   // 391
    const dim3 gemmGrdH(gemmGrdX, HID / 64);                  // N=256
    const dim3 gemmGrdC(gemmGrdX, CPAD / 64);                 // N=40 (masked)
    const size_t gemmLds = (size_t)(GEMM_K / 2) * LDS_PADP * sizeof(v2f);  // 80 KB

    // res0 = raw_x @ W_res0
    gemm_wmma_f32<<<gemmGrdH, gemmBlk, gemmLds, stream>>>(raw_x, wtRes, res0, M, HID);

    // three GCN layers: x = relu(spmm(A, x @ W) + res0)
    const long long n4H    = (long long)matElems / 4;
    const int zeroBlksH    = (int)((n4H + 255) / 256);
    const long long spmmTH = (long long)E * (HID / 4);
    const int spmmBlksH    = (int)((spmmTH + 255) / 256);

    const float* wt[3] = { wtGC0, wtGC1, wtGC2 };
    const float* xin = raw_x;
    for (int l = 0; l < 3; ++l) {
        gemm_wmma_f32<<<gemmGrdH, gemmBlk, gemmLds, stream>>>(xin, wt[l], bufA, M, HID);
        zero_f32x4<<<zeroBlksH, 256, 0, stream>>>((float4*)bufB, n4H);
        spmm_scatter<HID / 4><<<spmmBlksH, 256, 0, stream>>>(edge_row, edge_col, edge_val,
                                                             bufA, bufB, E);
        add_res_relu<<<zeroBlksH, 256, 0, stream>>>((float4*)bufB, (const float4*)res0, n4H);
        xin = bufB;
    }

    // final layer: y = spmm(A, x @ W_gc3) + res0 @ W_res_last
    const size_t yElems    = (size_t)M * C_OUT;   // 2M floats
    const long long n4C    = (long long)yElems / 4;
    const int zeroBlksC    = (int)((n4C + 255) / 256);
    const long long spmmTC = (long long)E * (C_OUT / 4);
    const int spmmBlksC    = (int)((spmmTC + 255) / 256);

    // xw3 = x3 @ W_gc3 -> bufA
    gemm_wmma_f32<<<gemmGrdC, gemmBlk, gemmLds, stream>>>(bufB, wtGC3, bufA, M, C_OUT);
    // yacc (bufB) = 0, then scatter xw3 into it
    zero_f32x4<<<zeroBlksC, 256, 0, stream>>>((float4*)bufB, n4C);
    spmm_scatter<C_OUT / 4><<<spmmBlksC, 256, 0, stream>>>(edge_row, edge_col, edge_val,
                                                           bufA, bufB, E);
    // yres = res0 @ W_res_last -> bufA
    gemm_wmma_f32<<<gemmGrdC, gemmBlk, gemmLds, stream>>>(res0, wtRL, bufA, M, C_OUT);

    // out = log_softmax(yacc + yres)
    const int lsBlks = (int)(((long long)M * 32 + 255) / 256);
    log_softmax_rows<<<lsBlks, 256, 0, stream>>>(bufB, bufA, (float*)d_out, M, C_OUT);
}